// InterpLnr_21629455302837
// MI455X (gfx1250) — compile-verified
//
#include <hip/hip_runtime.h>

// Problem constants (match reference)
#define BB       16
#define SS       65            // segments per batch
#define RR       (BB * SS)     // 1040 segment rows
#define WW       256           // candidate positions per segment
#define T_PAD    2176          // x.shape[1]
#define SEQ      2048          // output time length
#define DD       128           // feature dim
#define MIN_SEG  32

// ---------------------------------------------------------------------------
// Workspace layout (ints):
//   offs     [0      .. RR)        per-row exclusive segment offset
//   cnt      [RR     .. 2*RR)      per-row masked count c_r
//   rowStart [2*RR   .. 3*RR+1)    exclusive scan of cnt (rowStart[RR] = total)
//   total    [3*RR+1]              sum of cnt
// ---------------------------------------------------------------------------

// Kernel 1: per-batch exclusive cumsum of segment lengths. 1 block, BB threads.
__global__ void k_offsets(const int* __restrict__ len_seg_raw, int* __restrict__ offs) {
    int b = threadIdx.x;
    if (b < BB) {
        int acc = 0;
        for (int s = 0; s < SS; ++s) {
            offs[b * SS + s] = acc;
            acc += len_seg_raw[b * SS + s] + MIN_SEG;
        }
    }
}

// Kernel 2: per-row masked count, replicating the reference's f32 math exactly.
// Grid: RR blocks of WW threads.
__global__ void k_count(const float* __restrict__ scales,
                        const int* __restrict__ len_seg_raw,
                        const int* __restrict__ len_seq,
                        const int* __restrict__ offs,
                        int* __restrict__ cnt) {
    int r = blockIdx.x;
    int w = threadIdx.x;

    float sc  = scales[r] + 0.5f;
    float isc = (float)w / sc;          // idx_scaled
    float fl  = floorf(isc);            // idx_fl
    int   seg = len_seg_raw[r] + MIN_SEG;
    bool  m1  = fl < ((float)seg - 1.0f);
    float io  = fl + (float)offs[r];    // idx_org
    float lr  = (float)len_seq[r / SS];
    bool  m2  = io < (lr - 1.0f);

    __shared__ int red[WW];
    red[w] = (m1 && m2) ? 1 : 0;
    __syncthreads();
    for (int s2 = WW / 2; s2 > 0; s2 >>= 1) {
        if (w < s2) red[w] += red[w + s2];
        __syncthreads();
    }
    if (w == 0) cnt[r] = red[0];
}

// Kernel 3: exclusive Blelloch scan of cnt[RR] in LDS. 1 block, 1024 threads.
__global__ void k_scan(const int* __restrict__ cnt,
                       int* __restrict__ rowStart,
                       int* __restrict__ total) {
    __shared__ int sh[2048];
    int t  = threadIdx.x;
    int a0 = 2 * t, a1 = 2 * t + 1;
    sh[a0] = (a0 < RR) ? cnt[a0] : 0;
    sh[a1] = (a1 < RR) ? cnt[a1] : 0;

    // up-sweep
    for (int d = 1; d < 2048; d <<= 1) {
        __syncthreads();
        int i = (t + 1) * (d << 1) - 1;
        if (i < 2048) sh[i] += sh[i - d];
    }
    __syncthreads();
    if (t == 0) {
        int tot = sh[2047];
        total[0] = tot;
        rowStart[RR] = tot;
        sh[2047] = 0;
    }
    // down-sweep
    for (int d = 1024; d >= 1; d >>= 1) {
        __syncthreads();
        int i = (t + 1) * (d << 1) - 1;
        if (i < 2048) {
            int tmp = sh[i - d];
            sh[i - d] = sh[i];
            sh[i] += tmp;
        }
    }
    __syncthreads();
    if (a0 < RR) rowStart[a0] = sh[a0];
    if (a1 < RR) rowStart[a1] = sh[a1];
}

// Kernel 4: output gather + lerp. One wave32 per output row (b, t); each lane
// handles one float4 (32 lanes * 16B = 512B = full D row, global_*_b128).
// Grid: (BB*SEQ)/8 blocks of 256 threads (8 rows per block).
__global__ void k_gather(const float*  __restrict__ x,
                         const float*  __restrict__ scales,
                         const int*    __restrict__ offs,
                         const int*    __restrict__ rowStart,
                         const int*    __restrict__ total,
                         float*        __restrict__ out) {
    int lane = threadIdx.x & 31;
    int row  = blockIdx.x * 8 + (threadIdx.x >> 5);   // 0 .. BB*SEQ-1
    int b    = row >> 11;                             // / SEQ
    int t    = row & (SEQ - 1);

    float4* o = reinterpret_cast<float4*>(out + (size_t)row * DD) + lane;

    int tot = total[0];
    int L   = tot / BB;
    if (t >= L) {                       // zero-fill tail
        *o = make_float4(0.f, 0.f, 0.f, 0.f);
        return;
    }

    int g = b * L + t;                  // compacted slot (g < tot <= RR*WW)

    // find r with rowStart[r] <= g < rowStart[r+1]  (upper_bound - 1)
    int lo = 0, hi = RR;
    while (lo < hi) {
        int mid = (lo + hi) >> 1;
        if (rowStart[mid + 1] <= g) lo = mid + 1;
        else                        hi = mid;
    }
    int r = lo;
    int w = g - rowStart[r];            // mask is a prefix per row (monotone)

    float sc  = scales[r] + 0.5f;
    float isc = (float)w / sc;
    float fl  = floorf(isc);
    float lam = isc - fl;
    float io  = fl + (float)offs[r];

    int i_fl = (int)io;                 // io >= 0, integral
    if (i_fl > T_PAD - 1) i_fl = T_PAD - 1;
    if (i_fl < 0)         i_fl = 0;
    int i_cl = i_fl + 1;
    if (i_cl > T_PAD - 1) i_cl = T_PAD - 1;

    int bs = r / SS;                    // source batch from segment row

    const float4* pf =
        reinterpret_cast<const float4*>(x + ((size_t)bs * T_PAD + i_fl) * DD) + lane;
    const float4* pc =
        reinterpret_cast<const float4*>(x + ((size_t)bs * T_PAD + i_cl) * DD) + lane;

    float4 a = *pf;
    float4 c = *pc;
    float  om = 1.0f - lam;
    float4 y;
    y.x = om * a.x + lam * c.x;
    y.y = om * a.y + lam * c.y;
    y.z = om * a.z + lam * c.z;
    y.w = om * a.w + lam * c.w;
    *o = y;
}

extern "C" void kernel_launch(void* const* d_in, const int* in_sizes, int n_in,
                              void* d_out, int out_size, void* d_ws, size_t ws_size,
                              hipStream_t stream) {
    const float* x           = (const float*)d_in[0];  // (16, 2176, 128)
    const float* scales      = (const float*)d_in[1];  // (1040,)
    const int*   len_seq     = (const int*)d_in[2];    // (16,)
    const int*   len_seg_raw = (const int*)d_in[3];    // (1040, 1)
    float*       out         = (float*)d_out;          // (16, 2048, 128)

    int* wsi      = (int*)d_ws;
    int* offs     = wsi;                 // RR
    int* cnt      = wsi + RR;            // RR
    int* rowStart = wsi + 2 * RR;        // RR + 1
    int* total    = wsi + 3 * RR + 1;    // 1

    k_offsets<<<1, BB, 0, stream>>>(len_seg_raw, offs);
    k_count<<<RR, WW, 0, stream>>>(scales, len_seg_raw, len_seq, offs, cnt);
    k_scan<<<1, 1024, 0, stream>>>(cnt, rowStart, total);
    k_gather<<<(BB * SEQ) / 8, 256, 0, stream>>>(x, scales, offs, rowStart, total, out);
}